// GraphGen_12240656794041
// MI455X (gfx1250) — compile-verified
//
#include <hip/hip_runtime.h>

typedef __attribute__((ext_vector_type(2))) float v2f;
typedef __attribute__((ext_vector_type(8))) float v8f;

// ---------------------------------------------------------------------------
// Mean-aggregate gather: dst[r, :] = mean_i src[neigh[r, i], :]
// One block per output row, one thread per feature column (F == blockDim.x).
// ---------------------------------------------------------------------------
__global__ void gather_mean_kernel(const float* __restrict__ src,
                                   const int* __restrict__ neigh,
                                   float* __restrict__ dst,
                                   int S, int F) {
  __shared__ int idx[32];
  const int r = blockIdx.x;
  if ((int)threadIdx.x < S) idx[threadIdx.x] = neigh[(size_t)r * S + threadIdx.x];
  __syncthreads();
  const int c = threadIdx.x;
  float s = 0.0f;
#pragma unroll 4
  for (int i = 0; i < S; ++i)
    s += src[(size_t)idx[i] * F + c];
  dst[(size_t)r * F + c] = s * (1.0f / (float)S);
}

// ---------------------------------------------------------------------------
// Reparameterize: h = eps * exp(logvar) + mu   (mu/logvar already post-ReLU)
// ---------------------------------------------------------------------------
__global__ void reparam_kernel(const float* __restrict__ mu,
                               const float* __restrict__ logvar,
                               const float* __restrict__ eps,
                               float* __restrict__ h, int n) {
  int i = blockIdx.x * blockDim.x + threadIdx.x;
  if (i < n) h[i] = eps[i] * expf(logvar[i]) + mu[i];
}

// ---------------------------------------------------------------------------
// FP32 WMMA GEMM:  C[M,N] = act( A[M,K] @ B + bias )
//   BNK == 1 : B operand is W[N,K] row-major (computes A @ W^T)
//   BNK == 0 : B operand is B[K,N] row-major (computes A @ B)
// act: 0 = identity, 1 = relu, 2 = tanh.
// One wave -> (16*MT) x 64 output tile; MT*4 f32 accumulators, A fragments
// reused across the 4 N-subtiles, B fragments reused across the MT M-subtiles.
//
// V_WMMA_F32_16X16X4_F32 fragments (wave32, ISA 7.12.2):
//   A 16x4 : lane<16 rows M=0..15 K={0,1}; lane>=16 same rows K={2,3}
//   B 4x16 : VGPR0 = K=0 (lanes 0-15) / K=2 (lanes 16-31); VGPR1 = K=1 / K=3
//   C 16x16: VGPR i = M=i (lanes 0-15) / M=8+i (lanes 16-31), N = lane%16
// ---------------------------------------------------------------------------
template <int MT, int BNK>
__global__ void gemm_wmma_kernel(const float* __restrict__ A,
                                 const float* __restrict__ Bm,
                                 const float* __restrict__ bias,
                                 float* __restrict__ C,
                                 int M, int N, int K, int act) {
  const int lane = threadIdx.x & 31;
  const int wave = (int)((blockIdx.x * blockDim.x + threadIdx.x) >> 5);
  const int wpr = N >> 6;                       // 64-wide tiles per row of C
  const int ntiles = (M / (16 * MT)) * wpr;
  if (wave >= ntiles) return;
  const int m0 = (wave / wpr) * (16 * MT);
  const int n0 = (wave % wpr) << 6;
  const int hl = lane >> 4;                     // half-wave select (K pair)
  const int l16 = lane & 15;

  v8f acc[MT][4];
#pragma unroll
  for (int mt = 0; mt < MT; ++mt)
#pragma unroll
    for (int j = 0; j < 4; ++j) acc[mt][j] = (v8f){};

  const float* ap[MT];
#pragma unroll
  for (int mt = 0; mt < MT; ++mt)
    ap[mt] = A + (size_t)(m0 + 16 * mt + l16) * K + 2 * hl;

  if (BNK) {
    const float* bp[4];
#pragma unroll
    for (int j = 0; j < 4; ++j)
      bp[j] = Bm + (size_t)(n0 + 16 * j + l16) * K + 2 * hl;
    for (int k = 0; k < K; k += 4) {
      v2f a[MT], b[4];
#pragma unroll
      for (int mt = 0; mt < MT; ++mt) a[mt] = *(const v2f*)(ap[mt] + k);
#pragma unroll
      for (int j = 0; j < 4; ++j) b[j] = *(const v2f*)(bp[j] + k);
#pragma unroll
      for (int mt = 0; mt < MT; ++mt)
#pragma unroll
        for (int j = 0; j < 4; ++j)
          acc[mt][j] = __builtin_amdgcn_wmma_f32_16x16x4_f32(
              false, a[mt], false, b[j], (short)0, acc[mt][j], false, false);
    }
  } else {
    for (int k = 0; k < K; k += 4) {
      v2f a[MT], b[4];
#pragma unroll
      for (int mt = 0; mt < MT; ++mt) a[mt] = *(const v2f*)(ap[mt] + k);
      const float* r0 = Bm + (size_t)(k + 2 * hl) * N + n0 + l16;
      const float* r1 = r0 + N;
#pragma unroll
      for (int j = 0; j < 4; ++j) b[j] = (v2f){r0[16 * j], r1[16 * j]};
#pragma unroll
      for (int mt = 0; mt < MT; ++mt)
#pragma unroll
        for (int j = 0; j < 4; ++j)
          acc[mt][j] = __builtin_amdgcn_wmma_f32_16x16x4_f32(
              false, a[mt], false, b[j], (short)0, acc[mt][j], false, false);
    }
  }

  float bb[4] = {0.f, 0.f, 0.f, 0.f};
  if (bias) {
#pragma unroll
    for (int j = 0; j < 4; ++j) bb[j] = bias[n0 + 16 * j + l16];
  }
#pragma unroll
  for (int mt = 0; mt < MT; ++mt) {
#pragma unroll
    for (int i = 0; i < 8; ++i) {
      const int row = m0 + 16 * mt + i + 8 * hl;
      float* cr = C + (size_t)row * N + n0 + l16;
#pragma unroll
      for (int j = 0; j < 4; ++j) {
        float v = acc[mt][j][i] + bb[j];
        if (act == 1) v = fmaxf(v, 0.f);
        else if (act == 2) v = tanhf(v);
        cr[16 * j] = v;
      }
    }
  }
}

extern "C" void kernel_launch(void* const* d_in, const int* in_sizes, int n_in,
                              void* d_out, int out_size, void* d_ws, size_t ws_size,
                              hipStream_t stream) {
  (void)in_sizes; (void)n_in; (void)out_size; (void)ws_size;

  const float* features = (const float*)d_in[0];
  const int*   neigh1   = (const int*)d_in[1];
  const int*   neigh2   = (const int*)d_in[2];
  const float* eps      = (const float*)d_in[3];
  const float* W1       = (const float*)d_in[4];
  const float* W2       = (const float*)d_in[5];
  const float* W3       = (const float*)d_in[6];
  const float* Wd1      = (const float*)d_in[7];
  const float* b1       = (const float*)d_in[8];
  const float* Wd2      = (const float*)d_in[9];
  const float* b2       = (const float*)d_in[10];
  const float* M1       = (const float*)d_in[11];
  const float* M2       = (const float*)d_in[12];

  const int Nn = 50000, F = 256, S = 20, Bb = 8192;
  const int L1 = 256, L2 = 128, D1 = 256, D2 = 256;

  // Workspace layout (floats). agg1 region is recycled for the decoder stage.
  float* ws   = (float*)d_ws;
  float* agg1 = ws;                           // [50000, 256]
  float* h1   = ws + (size_t)Nn * F;          // [50000, 256]
  float* agg2 = ws;                           // [8192, 256]   (reuses agg1)
  float* hbuf = agg2 + (size_t)Bb * L1;       // [8192, 128]
  float* o1   = hbuf + (size_t)Bb * L2;       // [8192, 256]
  float* o2   = o1 + (size_t)Bb * D1;         // [8192, 256]
  float* e1   = o2 + (size_t)Bb * D2;         // [8192, 256]
  float* e2   = e1 + (size_t)Bb * D2;         // [8192, 256]

  float* out     = (float*)d_out;
  float* mu_out  = out;                           // [8192, 128]
  float* lv_out  = out + (size_t)Bb * L2;         // [8192, 128]
  float* adj_out = out + (size_t)2 * Bb * L2;     // [8192, 8192]

  auto launch_gemm = [&](const float* A, const float* Bm, const float* bias,
                         float* C, int M, int N, int K, int act, int bnk) {
    if (M % 32 == 0) {
      long long waves = (long long)(M / 32) * (N / 64);
      int blocks = (int)((waves * 32 + 255) / 256);
      if (bnk)
        gemm_wmma_kernel<2, 1><<<blocks, 256, 0, stream>>>(A, Bm, bias, C, M, N, K, act);
      else
        gemm_wmma_kernel<2, 0><<<blocks, 256, 0, stream>>>(A, Bm, bias, C, M, N, K, act);
    } else {
      long long waves = (long long)(M / 16) * (N / 64);
      int blocks = (int)((waves * 32 + 255) / 256);
      if (bnk)
        gemm_wmma_kernel<1, 1><<<blocks, 256, 0, stream>>>(A, Bm, bias, C, M, N, K, act);
      else
        gemm_wmma_kernel<1, 0><<<blocks, 256, 0, stream>>>(A, Bm, bias, C, M, N, K, act);
    }
  };

  // encode
  gather_mean_kernel<<<Nn, F, 0, stream>>>(features, neigh1, agg1, S, F);
  launch_gemm(agg1, W1, nullptr, h1, Nn, L1, F, /*relu*/1, 1);
  gather_mean_kernel<<<Bb, L1, 0, stream>>>(h1, neigh2, agg2, S, L1);
  launch_gemm(agg2, W2, nullptr, mu_out, Bb, L2, L1, /*relu*/1, 1);
  launch_gemm(agg2, W3, nullptr, lv_out, Bb, L2, L1, /*relu*/1, 1);

  // reparameterize
  const int nrep = Bb * L2;
  reparam_kernel<<<(nrep + 255) / 256, 256, 0, stream>>>(mu_out, lv_out, eps, hbuf, nrep);

  // decode
  launch_gemm(hbuf, Wd1, b1, o1, Bb, D1, L2, /*tanh*/2, 1);
  launch_gemm(o1, Wd2, b2, o2, Bb, D2, D1, /*id*/0, 1);
  launch_gemm(o2, M1, nullptr, e1, Bb, D2, D2, /*id*/0, /*B row-major KxN*/0);
  launch_gemm(o2, M2, nullptr, e2, Bb, D2, D2, /*id*/0, 0);

  // inner-product decoder: adj = e1 @ e2^T
  launch_gemm(e1, e2, nullptr, adj_out, Bb, Bb, D2, /*id*/0, 1);
}